// GAT_2516850835649
// MI455X (gfx1250) — compile-verified
//
#include <hip/hip_runtime.h>
#include <hip/hip_bf16.h>

typedef float v2f __attribute__((ext_vector_type(2)));
typedef float v8f __attribute__((ext_vector_type(8)));

#define KDIM 128
#define NEG_SLOPE 0.2f
#define EPS_F 1e-16f

// ---------------------------------------------------------------------------
// Dense GEMM  H[N,COUT] = X[N,KDIM] @ W[KDIM,COUT]  via V_WMMA_F32_16X16X4_F32
// 16 waves/block, each wave owns a 16-row tile and all COUT/16 col tiles.
// W is staged transposed in LDS (stride KDIM+2 to avoid bank conflicts).
// A-row index is clamped (not branched): garbage rows only feed D rows >= N,
// which are never stored -> branch-free, pipelinable global_load_b64 stream.
// Store tail has an unguarded fast path for fully in-range blocks.
// ---------------------------------------------------------------------------
template <int COUT>
__global__ __launch_bounds__(512) void gemm_wmma_f32(
    const float* __restrict__ X, const float* __restrict__ Wm,
    float* __restrict__ H, int N) {
  constexpr int LDW = KDIM + 2;            // pad: b64 stays 8B aligned, banks spread
  __shared__ float Ws[COUT * LDW];

  const int tid = threadIdx.x;
  // cooperative transpose load: Ws[c][k] = W[k][c]  (coalesced global reads)
  for (int idx = tid; idx < KDIM * COUT; idx += 512) {
    const int k = idx / COUT;
    const int c = idx % COUT;
    Ws[c * LDW + k] = Wm[idx];
  }
  __syncthreads();

  const int wave = tid >> 5;
  const int lane = tid & 31;
  const int i16  = lane & 15;
  const int half = lane >> 4;
  const int rowBase = blockIdx.x * 256 + wave * 16;
  const int rowA = rowBase + i16;          // A-matrix row this lane supplies
  const int rowAc = rowA < N ? rowA : (N - 1);   // clamp: branch-free loads

  v8f acc[COUT / 16] = {};

  const float* xrow = X + (long)rowAc * KDIM;
#pragma unroll 2
  for (int kk = 0; kk < KDIM / 4; ++kk) {
    const int k0 = kk * 4 + half * 2;      // lanes 0-15: K={0,1}, 16-31: K={2,3}
    const v2f a = *(const v2f*)(xrow + k0);
    // preload all B fragments first so ds waits overlap instead of serializing
    v2f bfr[COUT / 16];
#pragma unroll
    for (int ct = 0; ct < COUT / 16; ++ct) {
      // B fragment: lane -> {W[k0][col], W[k0+1][col]}, col = ct*16+i16
      bfr[ct] = *(const v2f*)(&Ws[(ct * 16 + i16) * LDW + k0]);
    }
#pragma unroll
    for (int ct = 0; ct < COUT / 16; ++ct) {
      acc[ct] = __builtin_amdgcn_wmma_f32_16x16x4_f32(
          false, a, false, bfr[ct], (short)0, acc[ct], false, false);
    }
  }

  // C/D layout: VGPR r holds rows {r, r+8} split across lane halves, col = i16
  float* hb = H + (long)rowBase * COUT + half * (8 * COUT) + i16;
  if (rowBase + 16 <= N) {
    // fast path: whole tile in range, straight-line clause-able store stream
#pragma unroll
    for (int ct = 0; ct < COUT / 16; ++ct) {
#pragma unroll
      for (int r = 0; r < 8; ++r) {
        hb[(long)r * COUT + ct * 16] = acc[ct][r];
      }
    }
  } else {
#pragma unroll
    for (int ct = 0; ct < COUT / 16; ++ct) {
#pragma unroll
      for (int r = 0; r < 8; ++r) {
        const int row = rowBase + r + half * 8;
        if (row < N) H[(long)row * COUT + ct * 16 + i16] = acc[ct][r];
      }
    }
  }
}

// ---------------------------------------------------------------------------
// Per-node attention logits: a_src[i] = dot(H[i,:], att_src), same for dst.
// One wave per node, butterfly shuffle reduction (wave32).
// ---------------------------------------------------------------------------
template <int C>
__global__ __launch_bounds__(256) void rowdot2(
    const float* __restrict__ H, const float* __restrict__ att_s,
    const float* __restrict__ att_d, float* __restrict__ out_s,
    float* __restrict__ out_d, int N) {
  const int wave = threadIdx.x >> 5;
  const int lane = threadIdx.x & 31;
  const int node = blockIdx.x * 8 + wave;
  if (node >= N) return;
  constexpr int PER = C / 32;
  const float* hp = H + (long)node * C + lane * PER;
  float s = 0.0f, d = 0.0f;
#pragma unroll
  for (int j = 0; j < PER; ++j) {
    const float hv = hp[j];
    s += hv * att_s[lane * PER + j];
    d += hv * att_d[lane * PER + j];
  }
#pragma unroll
  for (int off = 16; off >= 1; off >>= 1) {
    s += __shfl_xor(s, off, 32);
    d += __shfl_xor(d, off, 32);
  }
  if (lane == 0) { out_s[node] = s; out_d[node] = d; }
}

// monotonic order-preserving float<->uint encoding for atomicMax-based segmax
__device__ __forceinline__ unsigned enc_f(float f) {
  unsigned u = __float_as_uint(f);
  return (u & 0x80000000u) ? ~u : (u | 0x80000000u);
}
__device__ __forceinline__ float dec_f(unsigned u) {
  return __uint_as_float((u & 0x80000000u) ? (u ^ 0x80000000u) : ~u);
}

// pass A: e = leaky_relu(a_src[s] + a_dst[d]); segment max into m_enc[d]
__global__ __launch_bounds__(256) void edge_logit_max(
    const int* __restrict__ src, const int* __restrict__ dst,
    const float* __restrict__ a_s, const float* __restrict__ a_d,
    float* __restrict__ e_out, unsigned* __restrict__ m_enc, int E, int Esl) {
  const int e = blockIdx.x * 256 + threadIdx.x;
  if (e >= Esl) return;
  int s, d;
  if (e < E) { s = src[e]; d = dst[e]; } else { s = d = e - E; }
  float v = a_s[s] + a_d[d];
  v = (v >= 0.0f) ? v : NEG_SLOPE * v;
  e_out[e] = v;
  atomicMax(m_enc + d, enc_f(v));
}

// pass B: ex = exp(e - m[d]); segment sum into ssum[d]
__global__ __launch_bounds__(256) void edge_exp_sum(
    const int* __restrict__ dst, const unsigned* __restrict__ m_enc,
    float* __restrict__ e_io, float* __restrict__ ssum, int E, int Esl) {
  const int e = blockIdx.x * 256 + threadIdx.x;
  if (e >= Esl) return;
  const int d = (e < E) ? dst[e] : (e - E);
  const float ex = __expf(e_io[e] - dec_f(m_enc[d]));
  e_io[e] = ex;
  atomicAdd(ssum + d, ex);
}

// pass C: alpha = ex / (ssum[d] + eps)
__global__ __launch_bounds__(256) void edge_norm(
    const int* __restrict__ dst, const float* __restrict__ ssum,
    float* __restrict__ e_io, int E, int Esl) {
  const int e = blockIdx.x * 256 + threadIdx.x;
  if (e >= Esl) return;
  const int d = (e < E) ? dst[e] : (e - E);
  e_io[e] = e_io[e] / (ssum[d] + EPS_F);
}

// pass D: agg[d,:] += alpha * H[s,:]   (wave per edge; L2-resident atomics)
template <int C>
__global__ __launch_bounds__(256) void edge_aggregate(
    const int* __restrict__ src, const int* __restrict__ dst,
    const float* __restrict__ alpha, const float* __restrict__ H,
    float* __restrict__ agg, int E, long Esl) {
  const int wave = threadIdx.x >> 5;
  const int lane = threadIdx.x & 31;
  const long e = (long)blockIdx.x * 8 + wave;
  if (e >= Esl) return;
  int s, d;
  if (e < E) { s = src[e]; d = dst[e]; } else { s = d = (int)(e - E); }
  const float al = alpha[e];
  constexpr int PER = C / 32;
  const float* hp = H + (long)s * C + lane * PER;
  float* op = agg + (long)d * C + lane * PER;
#pragma unroll
  for (int j = 0; j < PER; ++j) atomicAdd(op + j, hp[j] * al);
}

// epilogue: out = agg + bias (optional relu)
__global__ __launch_bounds__(256) void bias_act(
    const float* __restrict__ a, const float* __restrict__ b,
    float* __restrict__ o, long n, int C, int do_relu) {
  const long i = (long)blockIdx.x * 256 + threadIdx.x;
  if (i >= n) return;
  float v = a[i] + b[(int)(i % C)];
  if (do_relu) v = fmaxf(v, 0.0f);
  o[i] = v;
}

extern "C" void kernel_launch(void* const* d_in, const int* in_sizes, int n_in,
                              void* d_out, int out_size, void* d_ws, size_t ws_size,
                              hipStream_t stream) {
  const float* x    = (const float*)d_in[0];
  const int*   ei   = (const int*)d_in[1];
  const float* W1   = (const float*)d_in[2];
  const float* as1  = (const float*)d_in[3];
  const float* ad1  = (const float*)d_in[4];
  const float* b1   = (const float*)d_in[5];
  const float* W2   = (const float*)d_in[6];
  const float* as2  = (const float*)d_in[7];
  const float* ad2  = (const float*)d_in[8];
  const float* b2   = (const float*)d_in[9];

  const int N   = in_sizes[0] / KDIM;   // 100000
  const int E   = in_sizes[1] / 2;      // 1600000
  const int Esl = E + N;                // with self-loops
  const int* src = ei;
  const int* dst = ei + E;

  // output layout: logits[N,64] | embedding[N,128] | alpha1[Esl] | alpha2[Esl]
  float* logits = (float*)d_out;
  float* emb    = logits + (long)N * 64;
  float* alpha1 = emb + (long)N * KDIM;
  float* alpha2 = alpha1 + Esl;

  // workspace: h_pre[N,128] | agg[N,128] | a_src[N] | a_dst[N] | m_enc[N] | ssum[N]
  float*    hpre  = (float*)d_ws;
  float*    agg   = hpre + (long)N * KDIM;
  float*    a_s   = agg + (long)N * KDIM;
  float*    a_d   = a_s + N;
  unsigned* m_enc = (unsigned*)(a_d + N);
  float*    ssum  = (float*)(m_enc + N);

  const int gemmBlocks = (N + 255) / 256;
  const int nodeBlocks = (N + 7) / 8;
  const int edgeBlocks = (Esl + 255) / 256;
  const int aggBlocks  = (Esl + 7) / 8;

  // ----------------- layer 1 -----------------
  gemm_wmma_f32<128><<<gemmBlocks, 512, 0, stream>>>(x, W1, hpre, N);
  rowdot2<128><<<nodeBlocks, 256, 0, stream>>>(hpre, as1, ad1, a_s, a_d, N);
  hipMemsetAsync(m_enc, 0, (size_t)N * 4, stream);           // enc-min sentinel
  hipMemsetAsync(ssum, 0, (size_t)N * 4, stream);
  hipMemsetAsync(agg, 0, (size_t)N * KDIM * 4, stream);
  edge_logit_max<<<edgeBlocks, 256, 0, stream>>>(src, dst, a_s, a_d, alpha1, m_enc, E, Esl);
  edge_exp_sum<<<edgeBlocks, 256, 0, stream>>>(dst, m_enc, alpha1, ssum, E, Esl);
  edge_norm<<<edgeBlocks, 256, 0, stream>>>(dst, ssum, alpha1, E, Esl);
  edge_aggregate<128><<<aggBlocks, 256, 0, stream>>>(src, dst, alpha1, hpre, agg, E, (long)Esl);
  {
    const long n = (long)N * KDIM;
    bias_act<<<(int)((n + 255) / 256), 256, 0, stream>>>(agg, b1, emb, n, KDIM, 1);
  }

  // ----------------- layer 2 (reuses workspace) -----------------
  gemm_wmma_f32<64><<<gemmBlocks, 512, 0, stream>>>(emb, W2, hpre, N);
  rowdot2<64><<<nodeBlocks, 256, 0, stream>>>(hpre, as2, ad2, a_s, a_d, N);
  hipMemsetAsync(m_enc, 0, (size_t)N * 4, stream);
  hipMemsetAsync(ssum, 0, (size_t)N * 4, stream);
  hipMemsetAsync(agg, 0, (size_t)N * 64 * 4, stream);
  edge_logit_max<<<edgeBlocks, 256, 0, stream>>>(src, dst, a_s, a_d, alpha2, m_enc, E, Esl);
  edge_exp_sum<<<edgeBlocks, 256, 0, stream>>>(dst, m_enc, alpha2, ssum, E, Esl);
  edge_norm<<<edgeBlocks, 256, 0, stream>>>(dst, ssum, alpha2, E, Esl);
  edge_aggregate<64><<<aggBlocks, 256, 0, stream>>>(src, dst, alpha2, hpre, agg, E, (long)Esl);
  {
    const long n = (long)N * 64;
    bias_act<<<(int)((n + 255) / 256), 256, 0, stream>>>(agg, b2, logits, n, 64, 0);
  }
}